// MeanBackflowSlater_19808389169440
// MI455X (gfx1250) — compile-verified
//
#include <hip/hip_runtime.h>
#include <math.h>

// Problem constants (match reference)
#define NS   512
#define NF   128
#define NB   16            // LU panel/block width
#define AH   2048          // ALPHA * NS hidden units
#define AHF  4096          // float view of complex hidden row

typedef float v2f __attribute__((ext_vector_type(2)));
typedef float v8f __attribute__((ext_vector_type(8)));

// ---------------------------------------------------------------------------
// Kernel 0: zero jastrow accumulators
// ---------------------------------------------------------------------------
__global__ void k_zero(float* p, int n) {
    int i = blockIdx.x * blockDim.x + threadIdx.x;
    if (i < n) p[i] = 0.0f;
}

// ---------------------------------------------------------------------------
// Kernel 1: per-row prep. 1 block (256 thr) per batch row.
//  - idx[b][0..127]: ascending occupied site indices (block prefix scan)
//  - h = tanh(n2 @ bf_W1 + b1) (complex 2-vec), sv = n2 @ rbm_vb (complex)
//  hsv layout per row: h0r,h0i,h1r,h1i,svr,svi
// ---------------------------------------------------------------------------
__global__ void k_prep(const float* __restrict__ n,
                       const float2* __restrict__ W1,   // (Ns,2) complex
                       const float2* __restrict__ b1,   // (2,) complex
                       const float2* __restrict__ vb,   // (Ns,) complex
                       float* __restrict__ hsv,
                       int* __restrict__ idxArr) {
    const int b = blockIdx.x;
    const int t = threadIdx.x;
    const float* nr = n + (size_t)b * NS;

    __shared__ int   sc[256];
    __shared__ float rs[256 * 6];

    // sites 2t, 2t+1 (contiguous -> order-preserving scan)
    int   s0 = 2 * t, s1 = 2 * t + 1;
    int   f0 = nr[s0] > 0.5f, f1 = nr[s1] > 0.5f;
    float g0 = f0 ? 1.0f : -1.0f, g1 = f1 ? 1.0f : -1.0f;

    // inclusive Hillis-Steele scan of per-thread counts
    sc[t] = f0 + f1;
    __syncthreads();
    for (int off = 1; off < 256; off <<= 1) {
        int v = (t >= off) ? sc[t - off] : 0;
        __syncthreads();
        sc[t] += v;
        __syncthreads();
    }
    int p = sc[t] - (f0 + f1);      // exclusive
    int* irow = idxArr + (size_t)b * NF;
    if (f0 && p < NF) irow[p++] = s0;
    if (f1 && p < NF) irow[p]   = s1;

    // partial sums: n2 @ W1 (2 cols) and n2 @ vb
    float a0r = 0, a0i = 0, a1r = 0, a1i = 0, avr = 0, avi = 0;
    {
        float2 w;
        w = W1[s0 * 2 + 0]; a0r += g0 * w.x; a0i += g0 * w.y;
        w = W1[s0 * 2 + 1]; a1r += g0 * w.x; a1i += g0 * w.y;
        w = vb[s0];         avr += g0 * w.x; avi += g0 * w.y;
        w = W1[s1 * 2 + 0]; a0r += g1 * w.x; a0i += g1 * w.y;
        w = W1[s1 * 2 + 1]; a1r += g1 * w.x; a1i += g1 * w.y;
        w = vb[s1];         avr += g1 * w.x; avi += g1 * w.y;
    }
    rs[t * 6 + 0] = a0r; rs[t * 6 + 1] = a0i;
    rs[t * 6 + 2] = a1r; rs[t * 6 + 3] = a1i;
    rs[t * 6 + 4] = avr; rs[t * 6 + 5] = avi;
    __syncthreads();
    for (int s = 128; s > 0; s >>= 1) {
        if (t < s)
            for (int c = 0; c < 6; ++c) rs[t * 6 + c] += rs[(t + s) * 6 + c];
        __syncthreads();
    }
    if (t == 0) {
        float z0r = rs[0] + b1[0].x, z0i = rs[1] + b1[0].y;
        float z1r = rs[2] + b1[1].x, z1i = rs[3] + b1[1].y;
        // tanh(x+iy) = (sinh2x + i sin2y)/(cosh2x + cos2y)
        float d0 = coshf(2.f * z0r) + cosf(2.f * z0i);
        float d1 = coshf(2.f * z1r) + cosf(2.f * z1i);
        float* o = hsv + (size_t)b * 6;
        o[0] = sinhf(2.f * z0r) / d0; o[1] = sinf(2.f * z0i) / d0;
        o[2] = sinhf(2.f * z1r) / d1; o[3] = sinf(2.f * z1i) / d1;
        o[4] = rs[4]; o[5] = rs[5];
    }
}

// ---------------------------------------------------------------------------
// Kernel 2: WMMA f32 GEMM (n2 @ rbm_W) fused with log_cosh reduction.
//  Block: 256 thr (8 waves). Tile: 128 batch rows x 16 hidden units.
//  B-tile deinterleaved re/im so complex pairs stay in-lane.
// ---------------------------------------------------------------------------
#define KC 64
#define APAD 66
__global__ void k_rbm(const float* __restrict__ n,
                      const float* __restrict__ Wf,   // rbm_W as float (Ns x 4096)
                      const float* __restrict__ hb,   // hidden bias as float (4096)
                      float* __restrict__ jas) {      // (B,2) accumulators
    __shared__ float As[128 * APAD];
    __shared__ float Bre[16 * APAD];
    __shared__ float Bim[16 * APAD];

    const int tid     = threadIdx.x;
    const int lane    = tid & 31;
    const int wave    = tid >> 5;
    const int rowBase = blockIdx.x * 128;
    const int hcBase  = blockIdx.y * 16;

    const int rowLocal = wave * 16 + (lane & 15);
    const int halfK    = (lane >> 4) * 2;

    v8f accRe = {0.f, 0.f, 0.f, 0.f, 0.f, 0.f, 0.f, 0.f};
    v8f accIm = {0.f, 0.f, 0.f, 0.f, 0.f, 0.f, 0.f, 0.f};

    for (int k0 = 0; k0 < NS; k0 += KC) {
        // stage A chunk (n2 = 2n-1), 128xKC
        for (int e = tid; e < 128 * KC; e += 256) {
            int r = e >> 6, kk = e & (KC - 1);
            As[r * APAD + kk] = 2.0f * n[(size_t)(rowBase + r) * NS + k0 + kk] - 1.0f;
        }
        // stage B chunk, deinterleave re/im, stored [h][kk]
        for (int e = tid; e < 16 * KC; e += 256) {
            int h = e >> 6, kk = e & (KC - 1);
            const float* src = Wf + (size_t)(k0 + kk) * AHF + 2 * (hcBase + h);
            Bre[h * APAD + kk] = src[0];
            Bim[h * APAD + kk] = src[1];
        }
        if (k0 + KC < NS)
            __builtin_prefetch(&n[(size_t)(rowBase + (tid >> 1)) * NS + k0 + KC], 0, 0);
        __syncthreads();

#pragma unroll
        for (int kk = 0; kk < KC; kk += 4) {
            v2f a  = *(const v2f*)&As[rowLocal * APAD + kk + halfK];
            v2f br = *(const v2f*)&Bre[(lane & 15) * APAD + kk + halfK];
            v2f bi = *(const v2f*)&Bim[(lane & 15) * APAD + kk + halfK];
            accRe = __builtin_amdgcn_wmma_f32_16x16x4_f32(false, a, false, br,
                                                          (short)0, accRe, false, false);
            accIm = __builtin_amdgcn_wmma_f32_16x16x4_f32(false, a, false, bi,
                                                          (short)0, accIm, false, false);
        }
        __syncthreads();
    }

    // epilogue: bias + complex log_cosh + per-row reduction
    const int h  = hcBase + (lane & 15);
    const float hbr = hb[2 * h], hbi = hb[2 * h + 1];
#pragma unroll
    for (int v = 0; v < 8; ++v) {
        float pr = accRe[v] + hbr;
        float pi = accIm[v] + hbi;
        float s  = (pr < 0.f) ? -1.f : 1.f;
        float x  = pr * s, y = pi * s;
        float e  = __expf(-2.f * x);
        float c2, s2;
        __sincosf(2.f * y, &s2, &c2);
        float wr = 0.5f + 0.5f * e * c2;
        float wi = -0.5f * e * s2;
        float lr = x + 0.5f * __logf(wr * wr + wi * wi);
        float li = y + atan2f(wi, wr);
        // reduce across the 16 hidden lanes of each half-wave
#pragma unroll
        for (int m = 1; m <= 8; m <<= 1) {
            lr += __shfl_xor(lr, m, 32);
            li += __shfl_xor(li, m, 32);
        }
        if ((lane & 15) == 0) {
            int row = rowBase + wave * 16 + v + ((lane >> 4) << 3);
            atomicAdd(&jas[2 * row + 0], lr);
            atomicAdd(&jas[2 * row + 1], li);
        }
    }
}

// ---------------------------------------------------------------------------
// Kernel 3: build phi (gathered backflow Slater matrix) in LDS, BLOCKED LU
// with partial pivoting.  Panel (NB=16) + U12 solve on VALU; trailing
// update A22 -= L21*U12 is a complex GEMM done with f32 WMMA (4 real WMMAs
// per complex MAC; L fragments negated in registers since F32 WMMA has no
// A/B NEG).  1 block (256 thr = 8 waves) per batch row.
// Dynamic LDS: 128 x 129 float2 (~132 KB).
// ---------------------------------------------------------------------------
#define MST 129
__device__ __forceinline__ float2 cmul(float2 a, float2 b) {
    return make_float2(a.x * b.x - a.y * b.y, a.x * b.y + a.y * b.x);
}

__global__ void k_lu(const float2* __restrict__ lam,  // (Nf, Ns)
                     const float2* __restrict__ W2,   // (2, Nf*Ns)
                     const float2* __restrict__ b2,   // (Nf*Ns,)
                     const float* __restrict__ hsv,
                     const int* __restrict__ idxArr,
                     const float* __restrict__ jas,
                     float* __restrict__ out) {
    extern __shared__ float2 M[];   // 128 x MST
    __shared__ float  redv[128];
    __shared__ int    redi[128];
    __shared__ float2 sInvP;
    __shared__ float  sLogAbs, sArg;
    __shared__ int    sPar, sPiv;

    const int b    = blockIdx.x;
    const int t    = threadIdx.x;
    const int lane = t & 31;
    const int wave = t >> 5;
    const int* irow = idxArr + (size_t)b * NF;
    const float* hv = hsv + (size_t)b * 6;
    const float2 h0 = make_float2(hv[0], hv[1]);
    const float2 h1 = make_float2(hv[2], hv[3]);

    if (t == 0) { sLogAbs = 0.f; sArg = 0.f; sPar = 0; }

    // build phi[i][j] = lam[i,c] * (h0*W2[0,i*Ns+c] + h1*W2[1,i*Ns+c] + b2[i*Ns+c])
    for (int e = t; e < NF * NF; e += 256) {
        int i = e >> 7, j = e & 127;
        int c = irow[j];
        size_t base = (size_t)i * NS + c;
        float2 w0 = W2[base];
        float2 w1 = W2[(size_t)NF * NS + base];
        float2 bb = b2[base];
        float2 bf = cmul(h0, w0);
        float2 t1 = cmul(h1, w1);
        bf.x += t1.x + bb.x; bf.y += t1.y + bb.y;
        M[i * MST + j] = cmul(lam[base], bf);
    }
    __syncthreads();

    for (int k0 = 0; k0 < NF; k0 += NB) {
        // ================= panel factorization (cols k0..k0+NB-1) ==========
        for (int kk = 0; kk < NB; ++kk) {
            const int k = k0 + kk;
            // pivot search over rows r >= k
            if (t < 128) {
                float bv = -1.f; int br = k;
                int r = k + t;
                if (r < NF) {
                    float2 a = M[r * MST + k];
                    bv = a.x * a.x + a.y * a.y; br = r;
                }
                redv[t] = bv; redi[t] = br;
            }
            __syncthreads();
            for (int s = 64; s > 0; s >>= 1) {
                if (t < s && redv[t + s] > redv[t]) {
                    redv[t] = redv[t + s]; redi[t] = redi[t + s];
                }
                __syncthreads();
            }
            if (t == 0) {
                int rp = redi[0];
                sPiv = rp;
                float2 pv = M[rp * MST + k];
                float m2 = pv.x * pv.x + pv.y * pv.y;
                sLogAbs += 0.5f * __logf(m2);
                sArg    += atan2f(pv.y, pv.x);
                if (rp != k) sPar ^= 1;
                float inv = 1.0f / m2;
                sInvP = make_float2(pv.x * inv, -pv.y * inv);
            }
            __syncthreads();
            const int rp = sPiv;
            if (rp != k) {
                for (int j = t; j < NF; j += 256) {
                    float2 tmp = M[k * MST + j];
                    M[k * MST + j]  = M[rp * MST + j];
                    M[rp * MST + j] = tmp;
                }
            }
            __syncthreads();
            const float2 invp = sInvP;
            for (int r = k + 1 + t; r < NF; r += 256)
                M[r * MST + k] = cmul(M[r * MST + k], invp);
            __syncthreads();
            // rank-1 update restricted to remaining panel columns
            const int w = k0 + NB - 1 - k;
            if (w > 0) {
                const int nrows = NF - 1 - k;
                const int total = nrows * w;
                for (int e = t; e < total; e += 256) {
                    int r = k + 1 + e / w;
                    int j = k + 1 + e % w;
                    float2 u = cmul(M[r * MST + k], M[k * MST + j]);
                    M[r * MST + j].x -= u.x;
                    M[r * MST + j].y -= u.y;
                }
            }
            __syncthreads();
        }

        const int nTrail = NF - k0 - NB;
        if (nTrail > 0) {
            // ============ U12 = L11^{-1} A12 (forward elimination) ==========
            for (int i = 0; i < NB - 1; ++i) {
                const int rows  = NB - 1 - i;
                const int total = rows * nTrail;
                for (int e = t; e < total; e += 256) {
                    int r = k0 + i + 1 + e / nTrail;
                    int j = k0 + NB + e % nTrail;
                    float2 u = cmul(M[r * MST + k0 + i], M[(k0 + i) * MST + j]);
                    M[r * MST + j].x -= u.x;
                    M[r * MST + j].y -= u.y;
                }
                __syncthreads();
            }
            // ============ trailing update A22 -= L21*U12 via WMMA ===========
            const int nt     = nTrail >> 4;          // 16x16 tiles per side
            const int cLane  = lane & 15;
            const int hi8    = (lane >> 4) << 3;
            const int halfK  = (lane >> 4) * 2;
            for (int tt = wave; tt < nt * nt; tt += 8) {
                const int rBase = k0 + NB + (tt / nt) * 16;
                const int jBase = k0 + NB + (tt % nt) * 16;
                const int jc = jBase + cLane;       // C/B column for this lane
                const int rA = rBase + cLane;       // A row for this lane
                v8f accRe, accIm;
#pragma unroll
                for (int v = 0; v < 8; ++v) {
                    float2 c = M[(rBase + v + hi8) * MST + jc];
                    accRe[v] = c.x; accIm[v] = c.y;
                }
#pragma unroll
                for (int m0 = 0; m0 < NB; m0 += 4) {
                    const int ka = k0 + m0 + halfK;
                    float2 a0 = M[rA * MST + ka];
                    float2 a1 = M[rA * MST + ka + 1];
                    float2 b0 = M[ka * MST + jc];
                    float2 b1 = M[(ka + 1) * MST + jc];
                    v2f naRe = { -a0.x, -a1.x };
                    v2f naIm = { -a0.y, -a1.y };
                    v2f aIm  = {  a0.y,  a1.y };
                    v2f bRe  = {  b0.x,  b1.x };
                    v2f bIm  = {  b0.y,  b1.y };
                    // Cre += (-Lre)*Ure + Lim*Uim ; Cim += (-Lre)*Uim + (-Lim)*Ure
                    accRe = __builtin_amdgcn_wmma_f32_16x16x4_f32(false, naRe, false, bRe,
                                                                  (short)0, accRe, false, false);
                    accRe = __builtin_amdgcn_wmma_f32_16x16x4_f32(false, aIm,  false, bIm,
                                                                  (short)0, accRe, false, false);
                    accIm = __builtin_amdgcn_wmma_f32_16x16x4_f32(false, naRe, false, bIm,
                                                                  (short)0, accIm, false, false);
                    accIm = __builtin_amdgcn_wmma_f32_16x16x4_f32(false, naIm, false, bRe,
                                                                  (short)0, accIm, false, false);
                }
#pragma unroll
                for (int v = 0; v < 8; ++v)
                    M[(rBase + v + hi8) * MST + jc] = make_float2(accRe[v], accIm[v]);
            }
            __syncthreads();
        }
    }

    if (t == 0) {
        const float TWO_PI = 6.28318530717958647f;
        float a = sArg + (sPar ? 3.14159265358979f : 0.f);
        a -= TWO_PI * rintf(a * (1.0f / TWO_PI));   // wrap to (-pi, pi]
        out[2 * b + 0] = sLogAbs + jas[2 * b + 0] + hv[4];
        out[2 * b + 1] = a       + jas[2 * b + 1] + hv[5];
    }
}

// ---------------------------------------------------------------------------
extern "C" void kernel_launch(void* const* d_in, const int* in_sizes, int n_in,
                              void* d_out, int out_size, void* d_ws, size_t ws_size,
                              hipStream_t stream) {
    const float*  n   = (const float*)d_in[0];
    const float2* lam = (const float2*)d_in[1];
    const float2* W1  = (const float2*)d_in[2];
    const float2* b1  = (const float2*)d_in[3];
    const float2* W2  = (const float2*)d_in[4];
    const float2* b2  = (const float2*)d_in[5];
    const float*  Wrb = (const float*)d_in[6];   // complex viewed as float pairs
    const float*  hb  = (const float*)d_in[7];
    const float2* vb  = (const float2*)d_in[8];

    const int B = in_sizes[0] / NS;

    // workspace layout
    float* jas = (float*)d_ws;          // 2*B floats
    float* hsv = jas + 2 * B;           // 6*B floats
    int*   idx = (int*)(hsv + 6 * B);   // B*NF ints

    k_zero<<<(2 * B + 255) / 256, 256, 0, stream>>>(jas, 2 * B);
    k_prep<<<B, 256, 0, stream>>>(n, W1, b1, vb, hsv, idx);
    dim3 g(B / 128, AH / 16);
    k_rbm<<<g, 256, 0, stream>>>(n, Wrb, hb, jas);
    size_t lds = (size_t)NF * MST * sizeof(float2);   // ~132 KB dynamic LDS
    k_lu<<<B, 256, lds, stream>>>(lam, W2, b2, hsv, idx, jas, (float*)d_out);
}